// SpectralLoss_79465484910837
// MI455X (gfx1250) — compile-verified
//
#include <hip/hip_runtime.h>
#include <hip/hip_bf16.h>
#include <math.h>

typedef float v2f __attribute__((ext_vector_type(2)));
typedef float v8f __attribute__((ext_vector_type(8)));

#define HW 512
#define NIMG 96          // B*C = 32*3
#define SSTR 257         // per-image bin-sum stride (bin 256 = discard slot)
#define LOSS_EPS 1e-8f
#define MPITCH 41        // LDS row pitch (floats); gcd(41,64)=1 -> conflict-free

__device__ __forceinline__ v8f wmma4(v2f a, v2f b, v8f c) {
  // D = A(16x4,f32) * B(4x16,f32) + C(16x16,f32)
  return __builtin_amdgcn_wmma_f32_16x16x4_f32(false, a, false, b, (short)0, c, false, false);
}

// ---------------------------------------------------------------------------
// Row-FFT kernel: 512-pt complex FFT of real rows, via 16x(FFT32) + twiddle +
// WMMA DFT-16.  Each wave handles 2 rows; WG = 128 threads = 4 waves = 8 rows.
// ---------------------------------------------------------------------------
__global__ __launch_bounds__(128) void spectral_rowfft(
    const float* __restrict__ src,   // full tensor base (NIMG images)
    float2* __restrict__ inter,      // chunk-local intermediate [cImgs][512][512]
    int imgBase)
{
  __shared__ float2 Wt[512];
  __shared__ float  MbRe[4][2][16][MPITCH];  // [wave][line][n1][k2] SoA
  __shared__ float  MbIm[4][2][16][MPITCH];

  const int tid  = threadIdx.x;
  const int wv   = tid >> 5;
  const int lane = tid & 31;
  const int hi   = lane >> 4;
  const int lo   = lane & 15;

  // W512[t] = exp(-2*pi*i*t/512)
  for (int t = tid; t < 512; t += 128) {
    float s, c;
    __sincosf(-6.28318530717958647692f * (float)t * (1.0f / 512.0f), &s, &c);
    Wt[t] = make_float2(c, s);
  }
  __syncthreads();

  // DFT-16 A-fragments: F[m][k] = W16^(m k) = W512[(m*k*32) & 511]
  // A layout (16x4 f32): lane m=lo; VGPR0 col = 4kk + (hi?2:0); VGPR1 = +1
  v2f FrA[4], FiA[4], FnA[4];
  #pragma unroll
  for (int kk = 0; kk < 4; ++kk) {
    const int c0 = 4 * kk + (hi ? 2 : 0);
    const float2 w0 = Wt[(lo * c0 * 32) & 511];
    const float2 w1 = Wt[(lo * (c0 + 1) * 32) & 511];
    v2f fr = {w0.x, w1.x};  FrA[kk] = fr;
    v2f fi = {w0.y, w1.y};  FiA[kk] = fi;
    v2f fn = {-w0.y, -w1.y}; FnA[kk] = fn;
  }

  constexpr int BR[32] = {0,16,8,24,4,20,12,28,2,18,10,26,6,22,14,30,
                          1,17,9,25,5,21,13,29,3,19,11,27,7,23,15,31};

  // this wave's two rows (chunk-local linear row ids)
  const int lrBase = blockIdx.x * 8 + wv * 2;
  const int lrL    = lrBase + hi;             // this lane's line
  const int imgL   = lrL >> 9;
  const int yL     = lrL & 511;
  const float* rp  = src + ((size_t)(imgBase + imgL) * HW + yL) * HW;

  // Step A: per-lane 32-pt FFT over n2 of x[n1 + 16*n2], n1 = lo
  float ar[32], ai[32];
  #pragma unroll
  for (int j = 0; j < 32; ++j) {
    ar[j] = __builtin_nontemporal_load(&rp[lo + 16 * BR[j]]);
    ai[j] = 0.0f;
  }
  #pragma unroll
  for (int s = 1; s <= 5; ++s) {
    const int h = 1 << (s - 1);
    const int sh = 9 - s;
    #pragma unroll
    for (int bt = 0; bt < 16; ++bt) {
      const int t  = bt & (h - 1);
      const int k  = (bt >> (s - 1)) << s;
      const int i0 = k + t, i1 = i0 + h;
      const float2 w = Wt[t << sh];
      const float br = w.x * ar[i1] - w.y * ai[i1];
      const float bi = w.x * ai[i1] + w.y * ar[i1];
      ar[i1] = ar[i0] - br; ai[i1] = ai[i0] - bi;
      ar[i0] += br;         ai[i0] += bi;
    }
  }
  // Step B: twiddle W512^(n1*k2)
  #pragma unroll
  for (int k2 = 0; k2 < 32; ++k2) {
    const float2 w = Wt[(lo * k2) & 511];
    const float r = w.x * ar[k2] - w.y * ai[k2];
    const float i = w.x * ai[k2] + w.y * ar[k2];
    ar[k2] = r; ai[k2] = i;
  }
  // stage to LDS (SoA)
  #pragma unroll
  for (int k2 = 0; k2 < 32; ++k2) {
    MbRe[wv][hi][lo][k2] = ar[k2];
    MbIm[wv][hi][lo][k2] = ai[k2];
  }
  __syncthreads();

  // Step C: X[k2 + 32*k1] = sum_n1 F16[k1][n1] * M[n1][k2]  (WMMA)
  #pragma unroll
  for (int l = 0; l < 2; ++l) {
    const int row  = lrBase + l;
    const int imgO = row >> 9;
    const int yO   = row & 511;
    float2* op = inter + ((size_t)imgO * HW + yO) * HW;
    #pragma unroll
    for (int p = 0; p < 2; ++p) {
      v8f Dr = {}; v8f Di = {};
      #pragma unroll
      for (int kk = 0; kk < 4; ++kk) {
        const int r0 = 4 * kk + (hi ? 2 : 0);
        const int ci = p * 16 + lo;
        v2f Br = {MbRe[wv][l][r0][ci], MbRe[wv][l][r0 + 1][ci]};
        v2f Bi = {MbIm[wv][l][r0][ci], MbIm[wv][l][r0 + 1][ci]};
        Dr = wmma4(FrA[kk], Br, Dr);
        Dr = wmma4(FnA[kk], Bi, Dr);
        Di = wmma4(FrA[kk], Bi, Di);
        Di = wmma4(FiA[kk], Br, Di);
      }
      #pragma unroll
      for (int j = 0; j < 8; ++j) {
        const int m  = j + hi * 8;
        const int kx = p * 16 + lo + 32 * m;
        op[kx] = make_float2(Dr[j], Di[j]);
      }
    }
  }
}

// ---------------------------------------------------------------------------
// Col-FFT kernel: 512-pt complex FFT down columns of the intermediate, then
// log-PSD + radial bin accumulation (LDS histogram -> per-image global sums).
// grid = (64 col-groups, chunkImgs); WG = 128 threads; 8 columns per WG.
// ---------------------------------------------------------------------------
__global__ __launch_bounds__(128) void spectral_colfft(
    const float2* __restrict__ inter,
    float* __restrict__ sums,        // [NIMG][SSTR]
    int imgBase)
{
  __shared__ float2 Wt[512];
  __shared__ float  MbRe[4][2][16][MPITCH];
  __shared__ float  MbIm[4][2][16][MPITCH];
  __shared__ float  binsLds[257];

  const int tid  = threadIdx.x;
  const int wv   = tid >> 5;
  const int lane = tid & 31;
  const int hi   = lane >> 4;
  const int lo   = lane & 15;

  for (int t = tid; t < 512; t += 128) {
    float s, c;
    __sincosf(-6.28318530717958647692f * (float)t * (1.0f / 512.0f), &s, &c);
    Wt[t] = make_float2(c, s);
  }
  for (int t = tid; t < 257; t += 128) binsLds[t] = 0.0f;
  __syncthreads();

  v2f FrA[4], FiA[4], FnA[4];
  #pragma unroll
  for (int kk = 0; kk < 4; ++kk) {
    const int c0 = 4 * kk + (hi ? 2 : 0);
    const float2 w0 = Wt[(lo * c0 * 32) & 511];
    const float2 w1 = Wt[(lo * (c0 + 1) * 32) & 511];
    v2f fr = {w0.x, w1.x};  FrA[kk] = fr;
    v2f fi = {w0.y, w1.y};  FiA[kk] = fi;
    v2f fn = {-w0.y, -w1.y}; FnA[kk] = fn;
  }

  constexpr int BR[32] = {0,16,8,24,4,20,12,28,2,18,10,26,6,22,14,30,
                          1,17,9,25,5,21,13,29,3,19,11,27,7,23,15,31};

  const int iL    = blockIdx.y;
  const int cBase = blockIdx.x * 8 + wv * 2;
  const int cLane = cBase + hi;
  const float2* cp = inter + (size_t)iL * HW * HW + cLane;

  float ar[32], ai[32];
  #pragma unroll
  for (int j = 0; j < 32; ++j) {
    const float2 v = cp[(size_t)(lo + 16 * BR[j]) * HW];
    ar[j] = v.x; ai[j] = v.y;
  }
  #pragma unroll
  for (int s = 1; s <= 5; ++s) {
    const int h = 1 << (s - 1);
    const int sh = 9 - s;
    #pragma unroll
    for (int bt = 0; bt < 16; ++bt) {
      const int t  = bt & (h - 1);
      const int k  = (bt >> (s - 1)) << s;
      const int i0 = k + t, i1 = i0 + h;
      const float2 w = Wt[t << sh];
      const float br = w.x * ar[i1] - w.y * ai[i1];
      const float bi = w.x * ai[i1] + w.y * ar[i1];
      ar[i1] = ar[i0] - br; ai[i1] = ai[i0] - bi;
      ar[i0] += br;         ai[i0] += bi;
    }
  }
  #pragma unroll
  for (int k2 = 0; k2 < 32; ++k2) {
    const float2 w = Wt[(lo * k2) & 511];
    const float r = w.x * ar[k2] - w.y * ai[k2];
    const float i = w.x * ai[k2] + w.y * ar[k2];
    ar[k2] = r; ai[k2] = i;
  }
  #pragma unroll
  for (int k2 = 0; k2 < 32; ++k2) {
    MbRe[wv][hi][lo][k2] = ar[k2];
    MbIm[wv][hi][lo][k2] = ai[k2];
  }
  __syncthreads();

  #pragma unroll
  for (int l = 0; l < 2; ++l) {
    const int   c  = cBase + l;
    const float dx = (float)(c - 256);
    #pragma unroll
    for (int p = 0; p < 2; ++p) {
      v8f Dr = {}; v8f Di = {};
      #pragma unroll
      for (int kk = 0; kk < 4; ++kk) {
        const int r0 = 4 * kk + (hi ? 2 : 0);
        const int ci = p * 16 + lo;
        v2f Br = {MbRe[wv][l][r0][ci], MbRe[wv][l][r0 + 1][ci]};
        v2f Bi = {MbIm[wv][l][r0][ci], MbIm[wv][l][r0 + 1][ci]};
        Dr = wmma4(FrA[kk], Br, Dr);
        Dr = wmma4(FnA[kk], Bi, Dr);
        Di = wmma4(FrA[kk], Bi, Di);
        Di = wmma4(FiA[kk], Br, Di);
      }
      #pragma unroll
      for (int j = 0; j < 8; ++j) {
        const int   m   = j + hi * 8;
        const int   ky  = p * 16 + lo + 32 * m;
        const float dy  = (float)(ky - 256);
        const float rad = sqrtf(dy * dy + dx * dx);
        int b = (int)rad;            // floor, rad >= 0
        b = b > 256 ? 256 : b;
        const float ps = Dr[j] * Dr[j] + Di[j] * Di[j];
        atomicAdd(&binsLds[b], logf(ps + LOSS_EPS));
      }
    }
  }
  __syncthreads();
  const int iG = imgBase + iL;
  for (int t = tid; t < 256; t += 128)
    atomicAdd(&sums[(size_t)iG * SSTR + t], binsLds[t]);
}

// ---------------------------------------------------------------------------
__global__ void spectral_zero(float* p, int n) {
  const int i = blockIdx.x * 256 + threadIdx.x;
  if (i < n) p[i] = 0.0f;
}

__global__ void spectral_counts(float* counts) {
  const int idx = blockIdx.x * 256 + threadIdx.x;   // grid covers 512*512
  const int y = idx >> 9, x = idx & 511;
  const float dy = (float)(y - 256), dx = (float)(x - 256);
  int b = (int)sqrtf(dy * dy + dx * dx);
  b = b > 256 ? 256 : b;
  atomicAdd(&counts[b], 1.0f);
}

__global__ __launch_bounds__(256) void spectral_final(
    const float* __restrict__ sp, const float* __restrict__ st,
    const float* __restrict__ counts, float* __restrict__ out)
{
  __shared__ float red[256];
  const int b = threadIdx.x;          // bin 0..255
  const float cnt = counts[b];
  float acc = 0.0f;
  for (int i = 0; i < NIMG; ++i)
    acc += fabsf(sp[(size_t)i * SSTR + b] - st[(size_t)i * SSTR + b]);
  acc = (cnt > 0.0f) ? acc / cnt : 0.0f;
  red[b] = acc;
  __syncthreads();
  for (int s = 128; s > 0; s >>= 1) {
    if (b < s) red[b] += red[b + s];
    __syncthreads();
  }
  if (b == 0) out[0] = red[0] / (float)(NIMG * 256);
}

// ---------------------------------------------------------------------------
extern "C" void kernel_launch(void* const* d_in, const int* in_sizes, int n_in,
                              void* d_out, int out_size, void* d_ws, size_t ws_size,
                              hipStream_t stream) {
  (void)in_sizes; (void)n_in; (void)out_size;
  const float* pred = (const float*)d_in[0];
  const float* targ = (const float*)d_in[1];
  float* out = (float*)d_out;

  float* wsf    = (float*)d_ws;
  float* sumsP  = wsf;                         // NIMG*SSTR
  float* sumsT  = wsf + (size_t)NIMG * SSTR;   // NIMG*SSTR
  float* counts = wsf + (size_t)2 * NIMG * SSTR;
  const size_t headFloats = (size_t)2 * NIMG * SSTR + SSTR;
  const size_t interOff   = ((headFloats * sizeof(float)) + 255) & ~(size_t)255;
  float2* inter = (float2*)((char*)d_ws + interOff);

  const size_t bytesPerImg = (size_t)HW * HW * sizeof(float2);
  size_t avail = (ws_size > interOff) ? ws_size - interOff : 0;
  int chunk = (int)(avail / bytesPerImg);
  if (chunk > 16) chunk = 16;      // ~33.5 MB: fits comfortably in 192 MB L2
  if (chunk < 1)  chunk = 1;

  spectral_zero<<<dim3(((int)headFloats + 255) / 256), dim3(256), 0, stream>>>(
      wsf, (int)headFloats);
  spectral_counts<<<dim3(HW * HW / 256), dim3(256), 0, stream>>>(counts);

  for (int t = 0; t < 2; ++t) {
    const float* src = t ? targ : pred;
    float* sums = t ? sumsT : sumsP;
    for (int base = 0; base < NIMG; base += chunk) {
      const int c = (NIMG - base < chunk) ? (NIMG - base) : chunk;
      // rows: c*512, 8 rows per WG
      spectral_rowfft<<<dim3(c * 64), dim3(128), 0, stream>>>(src, inter, base);
      // cols: 8 cols per WG, per image
      spectral_colfft<<<dim3(64, c), dim3(128), 0, stream>>>(inter, sums, base);
    }
  }
  spectral_final<<<dim3(1), dim3(256), 0, stream>>>(sumsP, sumsT, counts, out);
}